// MultiHeadAttention_SelfKvCache_11003706212642
// MI455X (gfx1250) — compile-verified
//
#include <hip/hip_runtime.h>
#include <hip/hip_bf16.h>
#include <math.h>

typedef float v2f __attribute__((ext_vector_type(2)));
typedef float v4f __attribute__((ext_vector_type(4)));
typedef float v8f __attribute__((ext_vector_type(8)));

#define N_STATE   1280
#define N_HEAD    20
#define HEAD_DIM  64
#define N_CTX     32
#define CACHE     16384
#define SPLITS    64
#define SPLIT_LEN (CACHE / SPLITS)   /* 256 */
#define SCALE     0.125f             /* 64^-0.5 */
#define S_STRIDE  (SPLIT_LEN + 4)    /* 260: 260 % 64 == 4 -> conflict-free A-frag ds loads */

static __device__ __forceinline__ v8f wmma4(v2f a, v2f b, v8f c) {
  // D = A(16x4 f32) * B(4x16 f32) + C(16x16 f32)
  return __builtin_amdgcn_wmma_f32_16x16x4_f32(
      /*neg_a=*/false, a, /*neg_b=*/false, b,
      /*c_mod=*/(short)0, c, /*reuse_a=*/false, /*reuse_b=*/false);
}

// ---------------------------------------------------------------------------
// Dense projection: out[32,1280] = A[32,1280] @ W[1280,1280]^T (+ bias)
// One wave per 16-column tile; the wave computes both 16-row M tiles so each
// W fragment is loaded once. 320 K-steps of v_wmma_f32_16x16x4_f32 per tile.
// ---------------------------------------------------------------------------
__global__ __launch_bounds__(32) void proj_kernel(
    const float* __restrict__ A, const float* __restrict__ W,
    const float* __restrict__ bias, float* __restrict__ out) {
  const int lane = threadIdx.x;
  const int n0   = blockIdx.x * 16;
  const int half = lane >> 4;        // 0 | 1
  const int lid  = lane & 15;
  const int koff = half << 1;        // K sub-offset per A/B layout

  v8f acc0 = {}; v8f acc1 = {};
  const float* a0p = A + (size_t)lid * N_STATE + koff;
  const float* a1p = A + (size_t)(16 + lid) * N_STATE + koff;
  const float* bp  = W + (size_t)(n0 + lid) * N_STATE + koff;

  for (int c = 0; c < N_STATE; c += 4) {
    v2f a0 = *(const v2f*)(a0p + c);
    v2f a1 = *(const v2f*)(a1p + c);
    v2f b  = *(const v2f*)(bp + c);   // B[k,n] = W[n0+n, c+k]
    acc0 = wmma4(a0, b, acc0);
    acc1 = wmma4(a1, b, acc1);
  }

  const float bb    = bias ? bias[n0 + lid] : 0.0f;
  const int   mbase = half << 3;
  for (int r = 0; r < 8; ++r) {
    const int m = mbase + r;
    out[(size_t)m * N_STATE + n0 + lid]        = acc0[r] + bb;
    out[(size_t)(16 + m) * N_STATE + n0 + lid] = acc1[r] + bb;
  }
}

// ---------------------------------------------------------------------------
// Bulk cache copy (both K and V): float4 grid-stride. RT policy on purpose:
// attention re-reads this data immediately; 192MB L2 should absorb it.
// ---------------------------------------------------------------------------
__global__ __launch_bounds__(256) void copy2_kernel(
    const v4f* __restrict__ sk, v4f* __restrict__ dk,
    const v4f* __restrict__ sv, v4f* __restrict__ dv, size_t n4) {
  size_t i      = (size_t)blockIdx.x * blockDim.x + threadIdx.x;
  size_t stride = (size_t)gridDim.x * blockDim.x;
  for (; i < n4; i += stride) {
    dk[i] = sk[i];
    dv[i] = sv[i];
  }
}

// Scatter the 32 freshly projected K/V rows into the copied caches.
__global__ __launch_bounds__(320) void scatter_kernel(
    const float* __restrict__ kp, const float* __restrict__ vp,
    const int* __restrict__ pos, float* __restrict__ kc, float* __restrict__ vc) {
  const int i = blockIdx.x;          // which new token row
  const int p = pos[i];              // cache row
  const int c = threadIdx.x * 4;     // 320 threads * 4 = 1280
  *(v4f*)&kc[(size_t)p * N_STATE + c] = *(const v4f*)&kp[(size_t)i * N_STATE + c];
  *(v4f*)&vc[(size_t)p * N_STATE + c] = *(const v4f*)&vp[(size_t)i * N_STATE + c];
}

// ---------------------------------------------------------------------------
// Flash-decoding attention, split-K over the cache.
// Block = (split s, head h), 4 waves. Pass A: S = scale*Q*K^T + mask -> LDS.
// Cooperative softmax stats. Pass B: O_partial = exp(S-m) * V via WMMA.
// Emits partial (O, m, l) for the combine kernel.
// ---------------------------------------------------------------------------
__global__ __launch_bounds__(128) void attn_kernel(
    const float* __restrict__ q,  const float* __restrict__ kc,
    const float* __restrict__ vc, const float* __restrict__ mask,
    float* __restrict__ part_O, float* __restrict__ part_m,
    float* __restrict__ part_l) {
  __shared__ float S[N_CTX][S_STRIDE];        // ~33 KB
  __shared__ float redM[N_CTX][4];
  __shared__ float redL[N_CTX][4];
  __shared__ float rowM[N_CTX];

  const int s    = blockIdx.x;      // split index
  const int h    = blockIdx.y;      // head
  const int t    = threadIdx.x;
  const int w    = t >> 5;          // wave 0..3
  const int lane = t & 31;
  const int half = lane >> 4;
  const int lid  = lane & 15;
  const int r0   = s * SPLIT_LEN;   // first cache row of this split

  // Preload scaled Q fragments for both M tiles (16 K-steps of 4)
  v2f a0[16], a1[16];
  {
    const float* q0 = q + (size_t)lid * N_STATE + h * HEAD_DIM + (half << 1);
    const float* q1 = q + (size_t)(16 + lid) * N_STATE + h * HEAD_DIM + (half << 1);
    for (int i = 0; i < 16; ++i) {
      a0[i] = (*(const v2f*)(q0 + 4 * i)) * SCALE;
      a1[i] = (*(const v2f*)(q1 + 4 * i)) * SCALE;
    }
  }

  // ---- Pass A: wave w computes S columns [w*64, w*64+64) ----
  const int wbase = w * 64;
  for (int nt = 0; nt < 4; ++nt) {
    const int lcol0 = wbase + nt * 16;
    const int col0  = r0 + lcol0;
    v8f s0 = {}; v8f s1 = {};
    // B[k,n] = kc[(col0+n)*1280 + h*64 + c + k]
    const float* kp = kc + (size_t)(col0 + lid) * N_STATE + h * HEAD_DIM + (half << 1);
    for (int i = 0; i < 16; ++i) {
      v2f b = *(const v2f*)(kp + 4 * i);
      s0 = wmma4(a0[i], b, s0);
      s1 = wmma4(a1[i], b, s1);
    }
    const int cidx = lcol0 + lid;
    const int gcol = r0 + cidx;
    const int mb   = half << 3;
    for (int r = 0; r < 8; ++r) {
      const int m = mb + r;
      S[m][cidx]      = s0[r] + mask[(size_t)m * CACHE + gcol];
      S[16 + m][cidx] = s1[r] + mask[(size_t)(16 + m) * CACHE + gcol];
    }
  }
  __syncthreads();

  // ---- Softmax stats: 4 threads per row, 64 columns each ----
  {
    const int row = t >> 2, part = t & 3;
    const int c0  = part * (SPLIT_LEN / 4);
    float pm = -INFINITY;
    for (int c = 0; c < SPLIT_LEN / 4; ++c) pm = fmaxf(pm, S[row][c0 + c]);
    redM[row][part] = pm;
  }
  __syncthreads();
  if (t < N_CTX) {
    float m = fmaxf(fmaxf(redM[t][0], redM[t][1]), fmaxf(redM[t][2], redM[t][3]));
    rowM[t] = m;
  }
  __syncthreads();
  {
    const int row = t >> 2, part = t & 3;
    const int c0  = part * (SPLIT_LEN / 4);
    const float mv = rowM[row];
    float pl = 0.0f;
    for (int c = 0; c < SPLIT_LEN / 4; ++c) {
      float e = __expf(S[row][c0 + c] - mv);
      S[row][c0 + c] = e;                        // S now holds P = exp(S - m)
      pl += e;
    }
    redL[row][part] = pl;
  }
  __syncthreads();
  if (t < N_CTX) {
    const size_t idx = ((size_t)h * SPLITS + s) * N_CTX + t;
    part_m[idx] = rowM[t];
    part_l[idx] = redL[t][0] + redL[t][1] + redL[t][2] + redL[t][3];
  }
  __syncthreads();

  // ---- Pass B: wave w computes O[:, w*16 .. w*16+16) over K = SPLIT_LEN ----
  const int ncol = h * HEAD_DIM + w * 16 + lid;  // column into vc rows
  const int bk   = half << 1;
  v8f o0 = {}; v8f o1 = {};
  for (int kk = 0; kk < SPLIT_LEN; kk += 4) {
    v2f b;
    b.x = vc[(size_t)(r0 + kk + bk) * N_STATE + ncol];
    b.y = vc[(size_t)(r0 + kk + bk + 1) * N_STATE + ncol];
    v2f p0 = *(const v2f*)&S[lid][kk + bk];       // A-frag, M tile 0 (bank-conflict-free)
    v2f p1 = *(const v2f*)&S[16 + lid][kk + bk];  // A-frag, M tile 1
    o0 = wmma4(p0, b, o0);
    o1 = wmma4(p1, b, o1);
  }
  const size_t obase = ((size_t)h * SPLITS + s) * N_CTX * HEAD_DIM;
  const int mb = half << 3;
  for (int r = 0; r < 8; ++r) {
    const int m = mb + r;
    part_O[obase + (size_t)m * HEAD_DIM + w * 16 + lid]        = o0[r];
    part_O[obase + (size_t)(16 + m) * HEAD_DIM + w * 16 + lid] = o1[r];
  }
}

// ---------------------------------------------------------------------------
// Combine split partials: O = sum_s exp(m_s - M) O_s / sum_s exp(m_s - M) l_s
// Writes the [32,1280] attention result (head-interleaved) into ws.
// ---------------------------------------------------------------------------
__global__ __launch_bounds__(256) void combine_kernel(
    const float* __restrict__ part_O, const float* __restrict__ part_m,
    const float* __restrict__ part_l, float* __restrict__ oc) {
  __shared__ float coef[N_CTX][SPLITS];   // 8 KB
  const int h = blockIdx.x, t = threadIdx.x;
  if (t < N_CTX) {
    float M = -INFINITY;
    for (int sp = 0; sp < SPLITS; ++sp)
      M = fmaxf(M, part_m[((size_t)h * SPLITS + sp) * N_CTX + t]);
    float L = 0.0f;
    for (int sp = 0; sp < SPLITS; ++sp) {
      float e = __expf(part_m[((size_t)h * SPLITS + sp) * N_CTX + t] - M);
      coef[t][sp] = e;
      L += e * part_l[((size_t)h * SPLITS + sp) * N_CTX + t];
    }
    const float inv = 1.0f / L;
    for (int sp = 0; sp < SPLITS; ++sp) coef[t][sp] *= inv;
  }
  __syncthreads();
  for (int idx = t; idx < N_CTX * HEAD_DIM; idx += 256) {
    const int m = idx >> 6, d = idx & 63;
    float acc = 0.0f;
    for (int sp = 0; sp < SPLITS; ++sp)
      acc += coef[m][sp] * part_O[(((size_t)h * SPLITS + sp) * N_CTX + m) * HEAD_DIM + d];
    oc[(size_t)m * N_STATE + h * HEAD_DIM + d] = acc;
  }
}

// ---------------------------------------------------------------------------
extern "C" void kernel_launch(void* const* d_in, const int* in_sizes, int n_in,
                              void* d_out, int out_size, void* d_ws, size_t ws_size,
                              hipStream_t stream) {
  (void)in_sizes; (void)n_in; (void)out_size; (void)ws_size;
  const float* x       = (const float*)d_in[0];
  const float* k_cache = (const float*)d_in[1];
  const float* v_cache = (const float*)d_in[2];
  const int*   pos     = (const int*)d_in[3];
  const float* mask    = (const float*)d_in[4];
  const float* Wq      = (const float*)d_in[5];
  const float* bq      = (const float*)d_in[6];
  const float* Wk      = (const float*)d_in[7];
  const float* Wv      = (const float*)d_in[8];
  const float* bv      = (const float*)d_in[9];
  const float* Wout    = (const float*)d_in[10];
  const float* bout    = (const float*)d_in[11];

  // Output layout: out[32*1280] | kc[16384*1280] | vc[16384*1280]
  float* out = (float*)d_out;
  float* kc  = out + (size_t)N_CTX * N_STATE;
  float* vc  = kc + (size_t)CACHE * N_STATE;

  // Workspace layout (floats): q | k | v | o_comb | part_O | part_m | part_l
  float* ws     = (float*)d_ws;
  float* q      = ws;
  float* kp     = q + (size_t)N_CTX * N_STATE;
  float* vp     = kp + (size_t)N_CTX * N_STATE;
  float* oc     = vp + (size_t)N_CTX * N_STATE;
  float* part_O = oc + (size_t)N_CTX * N_STATE;                       // 20*64*32*64
  float* part_m = part_O + (size_t)N_HEAD * SPLITS * N_CTX * HEAD_DIM;
  float* part_l = part_m + (size_t)N_HEAD * SPLITS * N_CTX;

  // 1) QKV projections (WMMA f32)
  proj_kernel<<<dim3(N_STATE / 16), 32, 0, stream>>>(x, Wq, bq, q);
  proj_kernel<<<dim3(N_STATE / 16), 32, 0, stream>>>(x, Wk, nullptr, kp);
  proj_kernel<<<dim3(N_STATE / 16), 32, 0, stream>>>(x, Wv, bv, vp);

  // 2) Copy both caches into the output, then scatter the 32 updated rows
  copy2_kernel<<<2048, 256, 0, stream>>>((const v4f*)k_cache, (v4f*)kc,
                                         (const v4f*)v_cache, (v4f*)vc,
                                         (size_t)CACHE * N_STATE / 4);
  scatter_kernel<<<N_CTX, 320, 0, stream>>>(kp, vp, pos, kc, vc);

  // 3) Flash-decoding attention over the updated caches (reads d_out's kc/vc)
  attn_kernel<<<dim3(SPLITS, N_HEAD), 128, 0, stream>>>(q, kc, vc, mask,
                                                        part_O, part_m, part_l);

  // 4) Combine split partials, 5) output projection
  combine_kernel<<<N_HEAD, 256, 0, stream>>>(part_O, part_m, part_l, oc);
  proj_kernel<<<dim3(N_STATE / 16), 32, 0, stream>>>(oc, Wout, bout, out);
}